// GNN_28449863368912
// MI455X (gfx1250) — compile-verified
//
#include <hip/hip_runtime.h>

#define N_NODES  100000
#define IN_FEATS 1433
#define HID      16
#define OUT_F    7
#define N_EDGES  300000
#define KPAD     1436          // IN_FEATS padded to multiple of 4
#define CHUNK_K  360           // LDS chunk of W1 rows (360*16*4 = 23040 B)
#define M_TILES  (N_NODES / 16)  // 6250 exact

typedef __attribute__((ext_vector_type(2))) float v2f;
typedef __attribute__((ext_vector_type(8))) float v8f;

// ---------------- init: zero agg1, out = broadcast(b2) ----------------
__global__ void gcn_init_kernel(float* __restrict__ agg1, float* __restrict__ out,
                                const float* __restrict__ b2) {
    int i = blockIdx.x * blockDim.x + threadIdx.x;
    if (i < N_NODES * HID)   agg1[i] = 0.0f;
    if (i < N_NODES * OUT_F) out[i]  = b2[i % OUT_F];
}

// ---------------- GEMM1: Y[100000,16] = X[100000,1433] @ W1[1433,16] ----------------
// One wave per 16-row M tile; W1 staged to LDS in zero-padded K chunks.
__global__ __launch_bounds__(256)
void gcn_gemm1_kernel(const float* __restrict__ X, const float* __restrict__ W1,
                      float* __restrict__ Y) {
    __shared__ float w1s[CHUNK_K * HID];
    const int tid  = threadIdx.x;
    const int lane = tid & 31;
    const int wave = tid >> 5;
    const int half = lane >> 4;   // 0: K pair {0,1}; 1: K pair {2,3}
    const int mcol = lane & 15;   // A row within tile / B,N column

    const int  tile  = blockIdx.x * 8 + wave;
    const bool valid = (tile < M_TILES);
    const int  rowA  = (valid ? tile : 0) * 16 + mcol;
    const float* Xrow = X + (size_t)rowA * IN_FEATS;

    v8f c = {};
    for (int kBase = 0; kBase < KPAD; kBase += CHUNK_K) {
        const int kCount = (KPAD - kBase) < CHUNK_K ? (KPAD - kBase) : CHUNK_K;
        __syncthreads();
        for (int i = tid; i < kCount * HID; i += 256) {
            int g = kBase * HID + i;
            w1s[i] = (g < IN_FEATS * HID) ? W1[g] : 0.0f;   // zero-pad K tail
        }
        __syncthreads();
        for (int ks = 0; ks < kCount; ks += 4) {
            const int kk = kBase + ks + half * 2;
            v2f a, b;
            a.x = (kk     < IN_FEATS) ? Xrow[kk]     : 0.0f;
            a.y = (kk + 1 < IN_FEATS) ? Xrow[kk + 1] : 0.0f;
            const int kl = ks + half * 2;
            b.x = w1s[kl * HID + mcol];
            b.y = w1s[(kl + 1) * HID + mcol];
            c = __builtin_amdgcn_wmma_f32_16x16x4_f32(false, a, false, b,
                                                      (short)0, c, false, false);
        }
    }
    if (valid) {
        const int baseRow = tile * 16 + half * 8;
        #pragma unroll
        for (int r = 0; r < 8; ++r)
            Y[(size_t)(baseRow + r) * HID + mcol] = c[r];
    }
}

// ---------------- scatter1: agg1[dst] += Y[src], 16 floats/edge ----------------
__global__ void gcn_scatter16_kernel(const float* __restrict__ Y, const int* __restrict__ src,
                                     const int* __restrict__ dst, float* __restrict__ agg1) {
    int t = blockIdx.x * blockDim.x + threadIdx.x;
    if (t >= N_EDGES * HID) return;
    int e = t >> 4, f = t & 15;
    int s = src[e], d = dst[e];
    atomicAdd(&agg1[(size_t)d * HID + f], Y[(size_t)s * HID + f]);
}

// ---------------- GEMM2: Z[100000,8] = relu(agg1 + b1) @ W2pad[16,16] ----------------
__global__ __launch_bounds__(256)
void gcn_gemm2_kernel(const float* __restrict__ agg1, const float* __restrict__ b1,
                      const float* __restrict__ W2, float* __restrict__ Z) {
    __shared__ float w2s[16 * 16];
    __shared__ float b1s[16];
    const int tid  = threadIdx.x;
    const int lane = tid & 31;
    const int wave = tid >> 5;
    const int half = lane >> 4;
    const int mcol = lane & 15;

    if (tid < 16 * 16) {
        int kr = tid >> 4, n = tid & 15;
        w2s[tid] = (n < OUT_F) ? W2[kr * OUT_F + n] : 0.0f;   // pad N 7 -> 16
    }
    if (tid < 16) b1s[tid] = b1[tid];
    __syncthreads();

    const int  tile  = blockIdx.x * 8 + wave;
    const bool valid = (tile < M_TILES);
    const int  rowA  = (valid ? tile : 0) * 16 + mcol;
    const float* Arow = agg1 + (size_t)rowA * HID;

    v8f c = {};
    #pragma unroll
    for (int ks = 0; ks < HID; ks += 4) {
        const int kk = ks + half * 2;
        v2f a, b;
        float ax = Arow[kk]     + b1s[kk];
        float ay = Arow[kk + 1] + b1s[kk + 1];
        a.x = ax > 0.0f ? ax : 0.0f;      // fused bias + ReLU on layer-1 output
        a.y = ay > 0.0f ? ay : 0.0f;
        b.x = w2s[kk * 16 + mcol];
        b.y = w2s[(kk + 1) * 16 + mcol];
        c = __builtin_amdgcn_wmma_f32_16x16x4_f32(false, a, false, b,
                                                  (short)0, c, false, false);
    }
    if (valid && mcol < 8) {
        const int baseRow = tile * 16 + half * 8;
        #pragma unroll
        for (int r = 0; r < 8; ++r)
            Z[(size_t)(baseRow + r) * 8 + mcol] = c[r];
    }
}

// ---------------- scatter2: out[dst] += Z[src], 7 floats/edge ----------------
__global__ void gcn_scatter7_kernel(const float* __restrict__ Z, const int* __restrict__ src,
                                    const int* __restrict__ dst, float* __restrict__ out) {
    int t = blockIdx.x * blockDim.x + threadIdx.x;
    if (t >= N_EDGES * 8) return;
    int e = t >> 3, f = t & 7;
    if (f >= OUT_F) return;
    int s = src[e], d = dst[e];
    atomicAdd(&out[(size_t)d * OUT_F + f], Z[(size_t)s * 8 + f]);
}

extern "C" void kernel_launch(void* const* d_in, const int* in_sizes, int n_in,
                              void* d_out, int out_size, void* d_ws, size_t ws_size,
                              hipStream_t stream) {
    (void)in_sizes; (void)n_in; (void)out_size; (void)ws_size;
    const float* X   = (const float*)d_in[0];
    const int*   src = (const int*)  d_in[1];
    const int*   dst = (const int*)  d_in[2];
    const float* W1  = (const float*)d_in[3];
    const float* b1  = (const float*)d_in[4];
    const float* W2  = (const float*)d_in[5];
    const float* b2  = (const float*)d_in[6];
    float* out = (float*)d_out;

    char*  ws   = (char*)d_ws;
    float* Y    = (float*)ws;                                     // [N,16]  (6.4 MB)
    float* agg1 = (float*)(ws + (size_t)N_NODES * HID * sizeof(float)); // [N,16]
    float* Z    = Y;  // Y dead after scatter1; reuse for [N,8]

    const int B = 256;
    // init: covers max(N*HID, N*OUT_F) elements
    gcn_init_kernel<<<(N_NODES * HID + B - 1) / B, B, 0, stream>>>(agg1, out, b2);
    // layer 1 projection (dominant: 573 MB read of X, WMMA f32)
    gcn_gemm1_kernel<<<(M_TILES + 7) / 8, B, 0, stream>>>(X, W1, Y);
    // layer 1 aggregation
    gcn_scatter16_kernel<<<(N_EDGES * HID + B - 1) / B, B, 0, stream>>>(Y, src, dst, agg1);
    // layer 2 projection (bias+ReLU fused into A operand)
    gcn_gemm2_kernel<<<(M_TILES + 7) / 8, B, 0, stream>>>(agg1, b1, W2, Z);
    // layer 2 aggregation (+b2 already in out)
    gcn_scatter7_kernel<<<(N_EDGES * 8 + B - 1) / B, B, 0, stream>>>(Z, src, dst, out);
}